// SimplifyLoss_8899172238077
// MI455X (gfx1250) — compile-verified
//
#include <hip/hip_runtime.h>

typedef __attribute__((ext_vector_type(2))) float v2f;
typedef __attribute__((ext_vector_type(8))) float v8f;

#define NB    8
#define NPRED 2048
#define NGT   8192
#define FLT_BIG 3.402823466e+38f

// ---------------------------------------------------------------------------
// Kernel A: for each pred j, min over all gts of ||gt_i - pred_j||^2.
// Wave owns 16 preds (B matrix cols + C broadcast), streams gts as A rows.
// D[i][j] = (gt_i . gt_i) + (pred_j . pred_j) - 2 gt_i . pred_j  via one
// v_wmma_f32_16x16x4_f32 per 16x16 tile (K=4 homogeneous-coordinate trick).
// ---------------------------------------------------------------------------
__global__ void __launch_bounds__(256)
mins_over_gt_kernel(const float* __restrict__ preds,
                    const float* __restrict__ gts,
                    float* __restrict__ mins_over_gt) {
  const int wave = threadIdx.x >> 5;
  const int lane = threadIdx.x & 31;
  const int n    = lane & 15;        // column index within tile
  const int half = lane >> 4;        // 0: K=0,1 / rows 0-7   1: K=2,3 / rows 8-15
  const int b    = blockIdx.y;
  const int pred_base = blockIdx.x * 128 + wave * 16;

  // Wave-resident pred j = pred_base + n
  const float* pp3 = preds + ((size_t)b * NPRED + pred_base + n) * 3;
  const float px = pp3[0], py = pp3[1], pz = pp3[2];
  const float pnorm = px * px + py * py + pz * pz;

  // B col j = [-2px, -2py, -2pz, 1]; lane half selects (K0,K1) vs (K2,K3)
  v2f bm;
  bm.x = half ? (-2.0f * pz) : (-2.0f * px);
  bm.y = half ? 1.0f         : (-2.0f * py);

  // C[i][j] = |pred_j|^2 : column-broadcast, identical in all 8 accum VGPRs
  v8f c;
#pragma unroll
  for (int r = 0; r < 8; ++r) c[r] = pnorm;

  const float* gb = gts + (size_t)b * NGT * 3;
  float runmin = FLT_BIG;

  for (int g0 = 0; g0 < NGT; g0 += 16) {
    // A row m = gt g0+n : [gx, gy, gz, |g|^2]
    const float* gp = gb + (size_t)(g0 + n) * 3;
    const float gx = gp[0], gy = gp[1], gz = gp[2];
    const float gn = gx * gx + gy * gy + gz * gz;
    v2f a;
    a.x = half ? gz : gx;
    a.y = half ? gn : gy;

    v8f d = __builtin_amdgcn_wmma_f32_16x16x4_f32(
        /*neg_a=*/false, a, /*neg_b=*/false, bm,
        /*c_mod=*/(short)0, c, /*reuse_a=*/false, /*reuse_b=*/false);

    // min over this lane's 8 rows (M = half*8 + r) of column n
    float t = fminf(fminf(fminf(d[0], d[1]), fminf(d[2], d[3])),
                    fminf(fminf(d[4], d[5]), fminf(d[6], d[7])));
    runmin = fminf(runmin, t);
  }

  // combine row halves: lane n holds rows 0-7 min, lane n+16 rows 8-15 min
  runmin = fminf(runmin, __shfl_xor(runmin, 16, 32));
  if (half == 0)
    mins_over_gt[(size_t)b * NPRED + pred_base + n] = runmin;
}

// ---------------------------------------------------------------------------
// Kernel B: for each gt i, min over all preds. Roles swapped: wave owns 16
// gts as B cols, streams preds as A rows.
// ---------------------------------------------------------------------------
__global__ void __launch_bounds__(256)
mins_over_pred_kernel(const float* __restrict__ preds,
                      const float* __restrict__ gts,
                      float* __restrict__ mins_over_pred) {
  const int wave = threadIdx.x >> 5;
  const int lane = threadIdx.x & 31;
  const int n    = lane & 15;
  const int half = lane >> 4;
  const int b    = blockIdx.y;
  const int gt_base = blockIdx.x * 128 + wave * 16;

  const float* gp3 = gts + ((size_t)b * NGT + gt_base + n) * 3;
  const float gx = gp3[0], gy = gp3[1], gz = gp3[2];
  const float gnorm = gx * gx + gy * gy + gz * gz;

  v2f bm;
  bm.x = half ? (-2.0f * gz) : (-2.0f * gx);
  bm.y = half ? 1.0f         : (-2.0f * gy);

  v8f c;
#pragma unroll
  for (int r = 0; r < 8; ++r) c[r] = gnorm;

  const float* pb = preds + (size_t)b * NPRED * 3;
  float runmin = FLT_BIG;

  for (int p0 = 0; p0 < NPRED; p0 += 16) {
    const float* pp3 = pb + (size_t)(p0 + n) * 3;
    const float px = pp3[0], py = pp3[1], pz = pp3[2];
    const float pn = px * px + py * py + pz * pz;
    v2f a;
    a.x = half ? pz : px;
    a.y = half ? pn : py;

    v8f d = __builtin_amdgcn_wmma_f32_16x16x4_f32(
        false, a, false, bm, (short)0, c, false, false);

    float t = fminf(fminf(fminf(d[0], d[1]), fminf(d[2], d[3])),
                    fminf(fminf(d[4], d[5]), fminf(d[6], d[7])));
    runmin = fminf(runmin, t);
  }

  runmin = fminf(runmin, __shfl_xor(runmin, 16, 32));
  if (half == 0)
    mins_over_pred[(size_t)b * NGT + gt_base + n] = runmin;
}

// ---------------------------------------------------------------------------
// Kernel C: loss = mean(mins_over_gt) + mean(mins_over_pred)
//                + mean_b(max_j mins_over_gt[b,:])
// ---------------------------------------------------------------------------
__global__ void __launch_bounds__(256)
final_reduce_kernel(const float* __restrict__ mg,
                    const float* __restrict__ mp,
                    float* __restrict__ out) {
  __shared__ float red[256];
  const int tid = threadIdx.x;

  float s1 = 0.0f;
  for (int i = tid; i < NB * NPRED; i += 256) s1 += mg[i];
  red[tid] = s1;
  __syncthreads();
  for (int off = 128; off > 0; off >>= 1) {
    if (tid < off) red[tid] += red[tid + off];
    __syncthreads();
  }
  const float S1 = red[0];
  __syncthreads();

  float s2 = 0.0f;
  for (int i = tid; i < NB * NGT; i += 256) s2 += mp[i];
  red[tid] = s2;
  __syncthreads();
  for (int off = 128; off > 0; off >>= 1) {
    if (tid < off) red[tid] += red[tid + off];
    __syncthreads();
  }
  const float S2 = red[0];
  __syncthreads();

  float maxacc = 0.0f;
  for (int bb = 0; bb < NB; ++bb) {
    float mx = -FLT_BIG;
    for (int j = tid; j < NPRED; j += 256)
      mx = fmaxf(mx, mg[bb * NPRED + j]);
    red[tid] = mx;
    __syncthreads();
    for (int off = 128; off > 0; off >>= 1) {
      if (tid < off) red[tid] = fmaxf(red[tid], red[tid + off]);
      __syncthreads();
    }
    maxacc += red[0];
    __syncthreads();
  }

  if (tid == 0)
    out[0] = S1 / (float)(NB * NPRED) + S2 / (float)(NB * NGT) + maxacc / (float)NB;
}

extern "C" void kernel_launch(void* const* d_in, const int* in_sizes, int n_in,
                              void* d_out, int out_size, void* d_ws, size_t ws_size,
                              hipStream_t stream) {
  const float* preds = (const float*)d_in[0];  // [8, 2048, 3]
  const float* gts   = (const float*)d_in[1];  // [8, 8192, 3]
  float* mg = (float*)d_ws;                    // [8, 2048]
  float* mp = mg + (size_t)NB * NPRED;         // [8, 8192]

  mins_over_gt_kernel<<<dim3(NPRED / 128, NB), 256, 0, stream>>>(preds, gts, mg);
  mins_over_pred_kernel<<<dim3(NGT / 128, NB), 256, 0, stream>>>(preds, gts, mp);
  final_reduce_kernel<<<1, 256, 0, stream>>>(mg, mp, (float*)d_out);
}